// MaskedNorm_55516747268447
// MI455X (gfx1250) — compile-verified
//
#include <hip/hip_runtime.h>
#include <hip/hip_bf16.h>
#include <stdint.h>

// Problem constants (B, N, D from the reference)
#define B_SZ   64
#define N_SZ   8192
#define D_SZ   128
#define NSPLIT 32                            // N-splits for the partial reduction
#define ROWS_PER_BLK (N_SZ / NSPLIT)         // 256 rows per block
#define TILE_ROWS    32                      // rows per LDS tile
#define TILE_FLOATS  (TILE_ROWS * D_SZ)      // 4096 floats = 16 KB
#define NTILES       (ROWS_PER_BLK / TILE_ROWS) // 8
#define THREADS      128                     // 4 wave32s
#define ISSUES       (TILE_FLOATS / 4 / THREADS) // 8 b128 issues per thread per tile

// The async-LDS builtin is typed: (int4 AS(1)*, int4 AS(3)*, imm offset, imm cpol)
typedef int v4i __attribute__((vector_size(4 * sizeof(int))));
typedef __attribute__((address_space(1))) v4i as1_v4i;
typedef __attribute__((address_space(3))) v4i as3_v4i;
typedef float v4f __attribute__((ext_vector_type(4)));

// ---- CDNA5 async global->LDS copy (ASYNCcnt path) -------------------------
// Generic pointers to LDS hold the LDS byte-offset in bits [31:0] (aperture in
// high bits), so a truncating int->ptr cast yields a valid AS(3) pointer.
__device__ __forceinline__ void async_copy_b128(const float* gsrc, float* ldst) {
#if defined(__AMDGCN__) && __has_builtin(__builtin_amdgcn_global_load_async_to_lds_b128)
  __builtin_amdgcn_global_load_async_to_lds_b128(
      (as1_v4i*)(uintptr_t)gsrc,                 // global source (AS1), const dropped via int
      (as3_v4i*)(unsigned)(uintptr_t)ldst,       // LDS dest: low 32 bits = LDS offset
      /*offset=*/0, /*cpol=*/0);                 // RT hint: X should stay in L2 for pass 3
#else
  unsigned l = (unsigned)(uintptr_t)ldst;  // LDS byte offset
  asm volatile("global_load_async_to_lds_b128 %0, %1, off"
               :: "v"(l), "v"(gsrc)
               : "memory");
#endif
}

#if defined(__AMDGCN__) && __has_builtin(__builtin_amdgcn_s_wait_asynccnt)
#define WAIT_ASYNC(n) __builtin_amdgcn_s_wait_asynccnt(n)
#else
#define WAIT_ASYNC(n) asm volatile("s_wait_asynccnt %0" :: "i"(n) : "memory")
#endif

// Issue one tile (TILE_ROWS x 128 floats) of async b128 loads.
// Thread-flat index covers the tile contiguously -> fully coalesced 2 KB/round.
__device__ __forceinline__ void issue_tile(const float* g, float* l, int tid) {
#pragma unroll
  for (int i = 0; i < ISSUES; ++i) {
    const int off = (i * THREADS + tid) * 4;  // in floats (16 B per thread)
    async_copy_b128(g + off, l + off);
  }
}

// ---------------------------------------------------------------------------
// Kernel 1: per-(batch, split) masked partial sums of x and x^2 over 256 rows.
// Thread d owns feature d; double-buffered async LDS tiles hide HBM latency.
// ---------------------------------------------------------------------------
__global__ __launch_bounds__(THREADS)
void k_partial(const float* __restrict__ X, const unsigned char* __restrict__ mask,
               float* __restrict__ ps, float* __restrict__ psq, float* __restrict__ cnt) {
  __shared__ float tile[2][TILE_FLOATS];
  __shared__ float validf[ROWS_PER_BLK];

  const int tid  = threadIdx.x;
  const int s    = blockIdx.x;   // split
  const int b    = blockIdx.y;   // batch
  const int row0 = s * ROWS_PER_BLK;

  const float* Xb = X + ((size_t)b * N_SZ + row0) * D_SZ;
  const unsigned char* mb = mask + (size_t)b * N_SZ + row0;  // mask stored as 1-byte bool

  for (int r = tid; r < ROWS_PER_BLK; r += THREADS)
    validf[r] = mb[r] ? 0.0f : 1.0f;  // reference uses ~mask as "valid"
  __syncthreads();

  issue_tile(Xb, tile[0], tid);

  float sum = 0.0f, sq = 0.0f;
  for (int t = 0; t < NTILES; ++t) {
    const int buf = t & 1;
    if (t + 1 < NTILES) {
      issue_tile(Xb + (size_t)(t + 1) * TILE_FLOATS, tile[buf ^ 1], tid);
      WAIT_ASYNC(ISSUES);   // tile t's (older, in-order) loads are complete
    } else {
      WAIT_ASYNC(0);
    }
    __syncthreads();        // all waves' portions of tile t visible

    const float* tp = tile[buf];
#pragma unroll
    for (int r = 0; r < TILE_ROWS; ++r) {
      const float f = validf[t * TILE_ROWS + r];   // broadcast LDS read
      const float v = tp[r * D_SZ + tid] * f;      // bank-conflict-free (d % 64 distinct)
      sum += v;
      sq   = fmaf(v, v, sq);                       // f in {0,1} => (vf)^2 == v^2 f
    }
    __syncthreads();        // reads done before buffer is refilled
  }

  const size_t o = ((size_t)b * NSPLIT + s) * D_SZ + tid;
  ps[o]  = sum;
  psq[o] = sq;
  if (tid == 0) {
    float c = 0.0f;
    for (int r = 0; r < ROWS_PER_BLK; ++r) c += validf[r];
    cnt[(size_t)b * NSPLIT + s] = c;
  }
}

// ---------------------------------------------------------------------------
// Kernel 2: fold the 32 partials per (b, d); emit scale/shift for the affine.
//   scale = W[d]/std,  shift = b[d] - mean*scale   (unbiased var, ddof=1)
// ---------------------------------------------------------------------------
__global__ __launch_bounds__(THREADS)
void k_stats(const float* __restrict__ ps, const float* __restrict__ psq,
             const float* __restrict__ cnt, const float* __restrict__ W,
             const float* __restrict__ bias,
             float* __restrict__ scale, float* __restrict__ shift) {
  const int d = threadIdx.x;
  const int b = blockIdx.x;
  float sum = 0.0f, sq = 0.0f, c = 0.0f;
  for (int s = 0; s < NSPLIT; ++s) {
    const size_t o = (size_t)b * NSPLIT + s;
    sum += ps[o * D_SZ + d];
    sq  += psq[o * D_SZ + d];
    c   += cnt[o];
  }
  const float mean = sum / c;
  const float var  = (sq - c * mean * mean) / (c - 1.0f);
  const float rstd = 1.0f / sqrtf(var);
  const float sc   = rstd * W[d];
  scale[(size_t)b * D_SZ + d] = sc;
  shift[(size_t)b * D_SZ + d] = bias[d] - mean * sc;
}

// ---------------------------------------------------------------------------
// Kernel 3: normalize + affine; masked rows -> REPLACE*W + b == b[d].
// Block indices are REVERSED vs pass 1 so the earliest blocks re-read the
// X regions most recently cached in the 192 MB L2 (X is 256 MB: ~75% reuse).
// Output is written with non-temporal b128 stores to avoid evicting X.
// ---------------------------------------------------------------------------
__global__ __launch_bounds__(THREADS)
void k_norm(const float* __restrict__ X, const unsigned char* __restrict__ mask,
            const float* __restrict__ scale, const float* __restrict__ shift,
            const float* __restrict__ bias, float* __restrict__ out) {
  __shared__ float tile[2][TILE_FLOATS];
  __shared__ float validf[ROWS_PER_BLK];

  const int tid  = threadIdx.x;
  const int lane = tid & 31;
  const int wv   = tid >> 5;        // wave id 0..3 -> row phase within a 4-row group
  const int s    = (NSPLIT - 1) - blockIdx.x;   // reverse traversal for L2 reuse
  const int b    = (B_SZ  - 1) - blockIdx.y;
  const int row0 = s * ROWS_PER_BLK;

  const float* Xb = X   + ((size_t)b * N_SZ + row0) * D_SZ;
  float*       Ob = out + ((size_t)b * N_SZ + row0) * D_SZ;
  const unsigned char* mb = mask + (size_t)b * N_SZ + row0;

  for (int r = tid; r < ROWS_PER_BLK; r += THREADS)
    validf[r] = mb[r] ? 0.0f : 1.0f;

  const int fd = lane * 4;          // 4 consecutive features per lane
  const v4f sc4 = *(const v4f*)(scale + (size_t)b * D_SZ + fd);
  const v4f sh4 = *(const v4f*)(shift + (size_t)b * D_SZ + fd);
  const v4f bb4 = *(const v4f*)(bias + fd);
  __syncthreads();

  issue_tile(Xb, tile[0], tid);

  for (int t = 0; t < NTILES; ++t) {
    const int buf = t & 1;
    if (t + 1 < NTILES) {
      issue_tile(Xb + (size_t)(t + 1) * TILE_FLOATS, tile[buf ^ 1], tid);
      WAIT_ASYNC(ISSUES);
    } else {
      WAIT_ASYNC(0);
    }
    __syncthreads();

    const float* tp = tile[buf];
#pragma unroll
    for (int i = 0; i < TILE_ROWS / 4; ++i) {
      const int r = i * 4 + wv;                       // wave w handles rows ≡ w (mod 4)
      const float f = validf[t * TILE_ROWS + r];      // wave-uniform broadcast
      const v4f x = *(const v4f*)(tp + r * D_SZ + fd);  // ds_load_b128
      v4f o;
      if (f != 0.0f) {
        o.x = fmaf(x.x, sc4.x, sh4.x);
        o.y = fmaf(x.y, sc4.y, sh4.y);
        o.z = fmaf(x.z, sc4.z, sh4.z);
        o.w = fmaf(x.w, sc4.w, sh4.w);
      } else {
        o = bb4;                                      // REPLACE(=0)*W + b
      }
      // streaming store: output is never re-read, keep L2 for X
      __builtin_nontemporal_store(
          o, (v4f*)(Ob + (size_t)(t * TILE_ROWS + r) * D_SZ + fd));
    }
    __syncthreads();
  }
}

// ---------------------------------------------------------------------------
// Host-side launcher. Inputs (setup_inputs order): X f32[64*8192*128],
// mask bool[64*8192] (1 byte/elem), W f32[128], b f32[128].
// Workspace: 2*B*S*D + B*S + 2*B*D floats ≈ 2.1 MB.
// ---------------------------------------------------------------------------
extern "C" void kernel_launch(void* const* d_in, const int* in_sizes, int n_in,
                              void* d_out, int out_size, void* d_ws, size_t ws_size,
                              hipStream_t stream) {
  const float*         X    = (const float*)d_in[0];
  const unsigned char* mask = (const unsigned char*)d_in[1];
  const float*         W    = (const float*)d_in[2];
  const float*         bias = (const float*)d_in[3];
  float*               out  = (float*)d_out;

  float* ws    = (float*)d_ws;
  float* ps    = ws;                                   // B*NSPLIT*D
  float* psq   = ps  + (size_t)B_SZ * NSPLIT * D_SZ;   // B*NSPLIT*D
  float* cnt   = psq + (size_t)B_SZ * NSPLIT * D_SZ;   // B*NSPLIT
  float* scale = cnt + (size_t)B_SZ * NSPLIT;          // B*D
  float* shift = scale + (size_t)B_SZ * D_SZ;          // B*D

  const dim3 blk(THREADS);
  const dim3 grid1(NSPLIT, B_SZ);   // 2048 blocks: fills the WGPs

  k_partial<<<grid1, blk, 0, stream>>>(X, mask, ps, psq, cnt);
  k_stats<<<dim3(B_SZ), blk, 0, stream>>>(ps, psq, cnt, W, bias, scale, shift);
  k_norm<<<grid1, blk, 0, stream>>>(X, mask, scale, shift, bias, out);
}